// QuantumConv2d_62388694941854
// MI455X (gfx1250) — compile-verified
//
#include <hip/hip_runtime.h>
#include <hip/hip_bf16.h>

// 16-qubit statevector sim of the quantum conv "circuit", 64 patches.
// Plan:
//   qc_sim:    1 workgroup per patch, statevector lives in d_ws (L2-resident,
//              32MB total, 512KB/patch). In-place gate passes with
//              __syncthreads between them. Layer-1 Rz is skipped (diagonal
//              phase, invariant |amp|^2).
//   qc_reduce: <Z_q> as a 64x65536x16 f32 GEMM via V_WMMA_F32_16X16X4_F32.
//              A = probabilities (computed on the fly from amplitudes),
//              B = sign matrix generated from index bits (no memory).

#define NQ 16
#define NSTATES 65536   // 2^16
#define NPATCH 64
#define PI_F 3.14159265f

typedef __attribute__((ext_vector_type(2))) float v2f;
typedef __attribute__((ext_vector_type(8))) float v8f;

// ---------------------------------------------------------------- simulation
__global__ void __launch_bounds__(256)
qc_sim(const float* __restrict__ x, const float* __restrict__ params,
       float2* __restrict__ state) {
    const int p    = blockIdx.x;          // patch id 0..63
    const int tid  = threadIdx.x;         // 0..255
    const int bimg = p >> 2;              // image
    const int pi   = (p >> 1) & 1;        // patch row
    const int pj   = p & 1;               // patch col

    __shared__ float cs[2][NQ], sn[2][NQ];   // cos/sin of Ry half-angles
    __shared__ float rz0[NQ];                // layer-0 Rz angles
    __shared__ float rzsum;

    if (tid < 2 * NQ) {
        const int l = tid >> 4, q = tid & 15;
        // patch pixel (row-major 4x4 flatten): data[q] = x[b,0, pi*4+q/4, pj*4+q%4]
        const float pix = x[bimg * 64 + (pi * 4 + (q >> 2)) * 8 + (pj * 4 + (q & 3))];
        const float ang = pix * PI_F + params[l * 2 * NQ + q];
        float s, c;
        __sincosf(0.5f * ang, &s, &c);
        cs[l][q] = c;
        sn[l][q] = s;
    }
    if (tid < NQ) rz0[tid] = params[NQ + tid];   // layer 0 Rz params
    if (tid == 0) {
        float t = 0.f;
        for (int q = 0; q < NQ; ++q) t += params[NQ + q];
        rzsum = t;
    }

    float2* st = state + ((size_t)p << 16);

    // |0...0>
    for (int i = tid; i < NSTATES; i += 256)
        st[i] = make_float2(i == 0 ? 1.f : 0.f, 0.f);
    __syncthreads();

    for (int l = 0; l < 2; ++l) {
        // ---- Ry(ang_q) on each qubit: disjoint pairs, in-place ----
        for (int q = 0; q < NQ; ++q) {
            const int mask = 1 << q;
            const float c = cs[l][q], s = sn[l][q];
            for (int h = tid; h < NSTATES / 2; h += 256) {
                const int i0 = ((h & ~(mask - 1)) << 1) | (h & (mask - 1));
                const int i1 = i0 | mask;
                const float2 a0 = st[i0], a1 = st[i1];
                st[i0] = make_float2(c * a0.x - s * a1.x, c * a0.y - s * a1.y);
                st[i1] = make_float2(s * a0.x + c * a1.x, s * a0.y + c * a1.y);
            }
            __syncthreads();
        }
        // ---- CNOT ring: swap (ctrl=1,tgt=0) <-> (ctrl=1,tgt=1) ----
        for (int g = 0; g < NQ; ++g) {
            const int ctrl = (g < NQ - 1) ? g : NQ - 1;
            const int tgt  = (g < NQ - 1) ? g + 1 : 0;
            const int b1 = ctrl < tgt ? ctrl : tgt;
            const int b2 = ctrl < tgt ? tgt : ctrl;
            const int m1 = (1 << b1) - 1, m2 = (1 << b2) - 1;
            for (int h = tid; h < NSTATES / 4; h += 256) {
                const int t0 = (h & m1) | ((h & ~m1) << 1);     // insert 0 @ b1
                const int t1 = (t0 & m2) | ((t0 & ~m2) << 1);   // insert 0 @ b2
                const int i0 = t1 | (1 << ctrl);                 // ctrl=1, tgt=0
                const int i1 = i0 | (1 << tgt);
                const float2 tmp = st[i0];
                st[i0] = st[i1];
                st[i1] = tmp;
            }
            __syncthreads();
        }
        // ---- Rz diagonal (layer 0 only; layer 1 phases don't affect |amp|^2)
        if (l == 0) {
            for (int i = tid; i < NSTATES; i += 256) {
                // theta = sum_{set bits} a_q - 0.5 * sum a_q
                float th = -0.5f * rzsum;
                int bits = i;
                for (int q = 0; q < NQ; ++q) {
                    if (bits & 1) th += rz0[q];
                    bits >>= 1;
                }
                float sth, cth;
                __sincosf(th, &sth, &cth);
                const float2 a = st[i];
                st[i] = make_float2(a.x * cth - a.y * sth, a.x * sth + a.y * cth);
            }
            __syncthreads();
        }
    }
}

// ------------------------------------------------------------- WMMA reduction
// D(16 patches x 16 qubits) += A(16x4 probs) * B(4x16 signs), K=65536 in
// chunks of 4 via V_WMMA_F32_16X16X4_F32.  A layout (ISA 7.12.2):
//   lanes 0-15 : VGPR0=A[m=lane][K=0], VGPR1=A[m=lane][K=1]
//   lanes16-31 : VGPR0=A[m=l-16][K=2], VGPR1=A[m=l-16][K=3]
// B mirrors with rows K=0/1 (lanes 0-15) and K=2/3 (lanes 16-31), N=lane&15.
__global__ void __launch_bounds__(256)
qc_reduce(const float2* __restrict__ state, float* __restrict__ out) {
    const int tile = blockIdx.x;           // 4 tiles of 16 patches
    const int wave = threadIdx.x >> 5;     // 8 waves split the K range
    const int lane = threadIdx.x & 31;
    const int l16  = lane & 15;
    const int hi   = lane >> 4;            // 0 -> K0/K1, 1 -> K2/K3
    const int patch = tile * 16 + l16;     // M = l16
    const int q     = l16;                 // N = l16
    const size_t sbase = (size_t)patch << 16;

    v8f acc = {};
    const int iters = NSTATES / 4 / 8;     // 2048 K-chunks per wave
    const int kstart = wave * iters * 4;
    for (int it = 0; it < iters; ++it) {
        const int kb = kstart + it * 4;
        const int k0 = kb + hi * 2;        // even -> 16B-aligned float4 load
        const float4 amp = *reinterpret_cast<const float4*>(state + sbase + k0);
        v2f a;
        a.x = amp.x * amp.x + amp.y * amp.y;       // prob(k0)
        a.y = amp.z * amp.z + amp.w * amp.w;       // prob(k0+1)
        v2f b;
        b.x = ((k0 >> q) & 1) ? -1.0f : 1.0f;      // sign row K0/K2
        b.y = (((k0 + 1) >> q) & 1) ? -1.0f : 1.0f;// sign row K1/K3
        acc = __builtin_amdgcn_wmma_f32_16x16x4_f32(
            false, a, false, b, (short)0, acc, false, false);
    }

    // combine the 8 per-wave accumulators (identical lane layout) through LDS
    __shared__ float part[8][8][32];
    for (int r = 0; r < 8; ++r) part[wave][r][lane] = acc[r];
    __syncthreads();

    if (wave == 0) {
        for (int r = 0; r < 8; ++r) {
            float sum = 0.f;
            for (int w = 0; w < 8; ++w) sum += part[w][r][lane];
            // C/D layout: VGPR r -> M=r (lanes 0-15) / M=8+r (lanes 16-31)
            const int m  = (lane < 16) ? r : (8 + r);
            const int qq = lane & 15;
            const int p  = tile * 16 + m;
            const int bi = p >> 2, ij = p & 3;
            // out[b][q][i][j], flat = b*64 + q*4 + i*2 + j
            out[bi * 64 + qq * 4 + ij] = sum;
        }
    }
}

extern "C" void kernel_launch(void* const* d_in, const int* in_sizes, int n_in,
                              void* d_out, int out_size, void* d_ws, size_t ws_size,
                              hipStream_t stream) {
    const float* x      = (const float*)d_in[0];   // (16,1,8,8) float32
    const float* params = (const float*)d_in[1];   // (64,) float32
    float* out          = (float*)d_out;           // (16,16,2,2) float32
    float2* state       = (float2*)d_ws;           // 64 * 65536 * 8B = 32 MB

    qc_sim<<<NPATCH, 256, 0, stream>>>(x, params, state);
    qc_reduce<<<4, 256, 0, stream>>>(state, out);
}